// AMPS_22333829939475
// MI455X (gfx1250) — compile-verified
//
#include <hip/hip_runtime.h>

typedef float v2f __attribute__((ext_vector_type(2)));
typedef float v8f __attribute__((ext_vector_type(8)));
typedef int   v4i __attribute__((vector_size(16)));
typedef __attribute__((address_space(1))) v4i g_v4i;   // global 16B vector
typedef __attribute__((address_space(3))) v4i l_v4i;   // LDS 16B vector

#define NSITES 1024
#define BSZ    2048
#define CHUNK  64

#if defined(__HIP_DEVICE_COMPILE__) && \
    __has_builtin(__builtin_amdgcn_global_load_async_to_lds_b128) && \
    __has_builtin(__builtin_amdgcn_s_wait_asynccnt)
#define ASYNC_STAGE 1
#else
#define ASYNC_STAGE 0
#endif

// Site-0 contribution; also (re)initializes d_out deterministically each call.
__global__ void amps_site0_kernel(const float* __restrict__ data,
                                  const float* __restrict__ tensors,
                                  float* __restrict__ out) {
    int b = blockIdx.x * blockDim.x + threadIdx.x;
    if (b >= BSZ) return;
    float t0 = tensors[0];              // tensors[0,0,0,0,0]
    float t1 = tensors[1];              // tensors[0,0,0,0,1]
    float m  = fmaxf(t0, t1);
    float lse = m + __logf(__expf(t0 - m) + __expf(t1 - m));
    float s = data[(size_t)b * (NSITES * 2)];   // spin at site 0
    out[b] = (s > 0.5f ? t0 : t1) - lse;
}

// Copy nf4 float4's from global to LDS (async on gfx1250, sync fallback).
__device__ __forceinline__ void stage_chunk(const float* __restrict__ gsrc,
                                            float* __restrict__ ldst,
                                            int nf4, int t) {
#if ASYNC_STAGE
    for (int i = t; i < nf4; i += 256) {
        __builtin_amdgcn_global_load_async_to_lds_b128(
            (g_v4i*)(gsrc + (size_t)i * 4),
            (l_v4i*)(ldst + (size_t)i * 4),
            0, 0);
    }
#else
    for (int i = t; i < nf4; i += 256)
        ((float4*)ldst)[i] = ((const float4*)gsrc)[i];
#endif
}

__device__ __forceinline__ void wait_stage() {
#if ASYNC_STAGE
    __builtin_amdgcn_s_wait_asynccnt(0);
#endif
}

// One wave advances TWO independent 16-batch chains (32 batches) for one site n
// via V_WMMA_F32_16X16X4_F32; A = [M0^T; M1^T; M0^T; M1^T] shared by both chains.
__global__ __launch_bounds__(256) void amps_chain_kernel(
    const float* __restrict__ data,      // (BSZ, NSITES, 2)
    const float* __restrict__ tensors,   // (NSITES, NSITES, 4, 4, 2)
    float* __restrict__ out) {           // (BSZ)
    __shared__ float smem[2 * CHUNK * 32];   // double-buffered matrix tiles

    const int  lane = threadIdx.x & 31;
    const int  wave = threadIdx.x >> 5;
    const int  n    = (NSITES - 1) - (int)(blockIdx.x >> 3);  // 1023 .. 1
    const int  tile = blockIdx.x & 7;
    const int  row  = lane & 15;
    const int  K0   = (lane >> 4) << 1;          // K pair: 0 or 2
    const bool hiH  = lane >= 16;
    const int  base = (tile * 8 + wave) << 5;    // 32 batches per wave
    // A[m,k] = T[n,j, l=k, r=m&3, p=(m>>2)&1]; per-lane dword offset in a tile
    const int  aoff = K0 * 8 + (row & 3) * 2 + ((row >> 2) & 1);

    const float* tn  = tensors + (size_t)n * (NSITES * 32);      // row n
    const float* db0 = data + (size_t)(base + row) * (NSITES * 2);
    const float* db1 = db0 + (size_t)16 * (NSITES * 2);

    // init left vectors v0[k] = T[n,0,0,k,p]  (B-operand layout)
    float s0 = db0[0];
    float s1 = db1[0];
    int   p0 = (s0 > 0.5f) ? 0 : 1;
    int   p1 = (s1 > 0.5f) ? 0 : 1;
    v2f B0, B1;
    B0.x = tn[K0 * 2 + p0];  B0.y = tn[(K0 + 1) * 2 + p0];
    B1.x = tn[K0 * 2 + p1];  B1.y = tn[(K0 + 1) * 2 + p1];

    if (n > 1)   // prologue: stage first chunk
        stage_chunk(tn + 32, smem, min(CHUNK, n - 1) * 8, (int)threadIdx.x);

    int cur = 0;
    for (int jbase = 1; jbase < n; jbase += CHUNK) {
        int cnt = min(CHUNK, n - jbase);
        wait_stage();
        __syncthreads();
        int nb = jbase + CHUNK;
        if (nb < n)   // overlap: stage next chunk into the other buffer
            stage_chunk(tn + (size_t)nb * 32, smem + (cur ^ 1) * (CHUNK * 32),
                        min(CHUNK, n - nb) * 8, (int)threadIdx.x);

        const float* buf = smem + cur * (CHUNK * 32);
        float sa = db0[2 * jbase];
        float sb = db1[2 * jbase];
        for (int jj = 0; jj < cnt; ++jj) {
            float sa_n = db0[2 * (jbase + jj + 1)];  // <= 2*1023, in bounds
            float sb_n = db1[2 * (jbase + jj + 1)];
            v2f Av;
            Av.x = buf[jj * 32 + aoff];
            Av.y = buf[jj * 32 + aoff + 8];
            v8f C = {};
            v8f D0 = __builtin_amdgcn_wmma_f32_16x16x4_f32(
                false, Av, false, B0, (short)0, C, false, false);
            v8f D1 = __builtin_amdgcn_wmma_f32_16x16x4_f32(
                false, Av, false, B1, (short)0, C, false, false);
            // per-lane repack D -> next B operands, phys chosen by spin
            bool q0 = !(sa > 0.5f);
            bool q1 = !(sb > 0.5f);
            float a0 = hiH ? D0[2] : D0[0];
            float a1 = hiH ? D0[6] : D0[4];
            float c0 = hiH ? D0[3] : D0[1];
            float c1 = hiH ? D0[7] : D0[5];
            B0.x = q0 ? a1 : a0;
            B0.y = q0 ? c1 : c0;
            float e0 = hiH ? D1[2] : D1[0];
            float e1 = hiH ? D1[6] : D1[4];
            float f0 = hiH ? D1[3] : D1[1];
            float f1 = hiH ? D1[7] : D1[5];
            B1.x = q1 ? e1 : e0;
            B1.y = q1 ? f1 : f0;
            sa = sa_n;
            sb = sb_n;
        }
        cur ^= 1;
    }

    // readout: logits_i = sum_k v[k] * T[n,n,k,0,i] via one WMMA per chain
    const float* td = tn + (size_t)n * 32;       // diagonal tile (n,n)
    int   rr = (row < 2) ? row : 1;
    float r0 = td[K0 * 8 + rr];
    float r1 = td[(K0 + 1) * 8 + rr];
    v2f Ar;
    Ar.x = (row < 2) ? r0 : 0.0f;
    Ar.y = (row < 2) ? r1 : 0.0f;
    v8f C2 = {};
    v8f E0 = __builtin_amdgcn_wmma_f32_16x16x4_f32(
        false, Ar, false, B0, (short)0, C2, false, false);
    v8f E1 = __builtin_amdgcn_wmma_f32_16x16x4_f32(
        false, Ar, false, B1, (short)0, C2, false, false);

    if (!hiH) {
        // chain 0 -> batch base+row
        float l0 = E0[0], l1 = E0[1];
        float sn = db0[2 * n];
        float m  = fmaxf(l0, l1);
        float lse = m + __logf(__expf(l0 - m) + __expf(l1 - m));
        atomicAdd(&out[base + row], (sn > 0.5f ? l0 : l1) - lse);
        // chain 1 -> batch base+16+row
        l0 = E1[0];  l1 = E1[1];
        sn = db1[2 * n];
        m   = fmaxf(l0, l1);
        lse = m + __logf(__expf(l0 - m) + __expf(l1 - m));
        atomicAdd(&out[base + 16 + row], (sn > 0.5f ? l0 : l1) - lse);
    }
}

extern "C" void kernel_launch(void* const* d_in, const int* in_sizes, int n_in,
                              void* d_out, int out_size, void* d_ws, size_t ws_size,
                              hipStream_t stream) {
    (void)in_sizes; (void)n_in; (void)out_size; (void)d_ws; (void)ws_size;
    const float* data    = (const float*)d_in[0];
    const float* tensors = (const float*)d_in[1];
    float* out = (float*)d_out;

    amps_site0_kernel<<<(BSZ + 255) / 256, 256, 0, stream>>>(data, tensors, out);
    amps_chain_kernel<<<(NSITES - 1) * 8, 256, 0, stream>>>(data, tensors, out);
}